// HopfONN_9070970929354
// MI455X (gfx1250) — compile-verified
//
#include <hip/hip_runtime.h>

// CDNA5 / gfx1250: wave32, WMMA f32 16x16x4.
typedef __attribute__((ext_vector_type(2))) float v2f;
typedef __attribute__((ext_vector_type(8))) float v8f;

#define SWAPX16_IMM 0x401F  // ds_swizzle group-of-32: xor=0x10, or=0, and=0x1f

__device__ __forceinline__ float swz16(float x) {
    return __builtin_bit_cast(float,
        __builtin_amdgcn_ds_swizzle(__builtin_bit_cast(int, x), SWAPX16_IMM));
}

// One wave handles 16 boards.
// State layout (== WMMA C/D layout, D = Out^T with M=osc, N=board):
//   lane l: board = l&15, oscillators (l<16 ? 0..7 : 8..15) in slots 0..7.
// A operands = W rows (M=osc_out, K=osc_in), preloaded, 4 K-chunks of v2f.
// B operands = Z^T (K=osc_in, N=board), rebuilt from state each dyn eval via
//   8 SWAPX16 swizzles + selects per matrix half.
__global__ __launch_bounds__(256) void hopf_ode_head_kernel(
    const int*   __restrict__ board,
    const float* __restrict__ W_re,
    const float* __restrict__ W_im,
    const float* __restrict__ head_w,
    const float* __restrict__ head_b,
    float*       __restrict__ out,
    int nboards)
{
    const int lane  = threadIdx.x & 31;
    const int gwave = (int)(((unsigned)blockIdx.x * blockDim.x + threadIdx.x) >> 5);
    const int b0    = gwave << 4;               // first board of this wave's tile
    if (b0 >= nboards) return;
    const int  col = lane & 15;                 // board column / W row (A: M = lane%16)
    const int  hi  = lane >> 4;                 // 0: osc 0-7, 1: osc 8-15
    const bool lo  = (hi == 0);

    // ---- Preload W into A layout: chunk kc, VGPR v -> W[M = col][K = 4kc + v + 2*hi]
    v2f Are[4], Aim[4], Anim[4];
#pragma unroll
    for (int kc = 0; kc < 4; ++kc) {
        const int j0 = 4 * kc + 2 * hi;
        const float* pr = W_re + col * 16 + j0;
        const float* pi = W_im + col * 16 + j0;
        v2f ar = { pr[0], pr[1] };
        v2f ai = { pi[0], pi[1] };
        Are[kc]  = ar;
        Aim[kc]  = ai;
        Anim[kc] = -ai;                         // for  -W_im * z_im  (f32 WMMA has no A/B neg)
    }

    // ---- Initial state: z0 = exp(i * 2pi * log2(clamp(board,1))/11)
    float zr[8], zi[8];
    {
        const int* bp = board + (size_t)(b0 + col) * 16 + 8 * hi;
#pragma unroll
        for (int s = 0; s < 8; ++s) {
            float v  = (float)bp[s];
            float x  = __log2f(fmaxf(v, 1.0f)) * (1.0f / 11.0f);
            float th = x * 6.28318530717958647692f;
            zr[s] = __cosf(th);
            zi[s] = __sinf(th);
        }
    }

    const float hstep = (0.5f * 20.0f) / 19.0f;   // h = DT*STEPS/(STEPS-1)

    // dyn(x) = (1 - |x|^2 + i)*x + W x   ; elementwise term feeds the WMMA C operand.
    auto dyn = [&](const float* xr, const float* xi, float* kr, float* ki) {
        v8f accR, accI;
#pragma unroll
        for (int s = 0; s < 8; ++s) {
            float m2 = fmaf(xr[s], xr[s], xi[s] * xi[s]);
            float a  = 1.0f - m2;                 // ALPHA - |z|^2
            accR[s]  = fmaf(a, xr[s], -xi[s]);    // re: (a)*xr - xi
            accI[s]  = fmaf(a, xi[s],  xr[s]);    // im: (a)*xi + xr
        }
        // State layout -> B layout (N = board in lane%16; K split low/high lane half)
        v2f Bre[4], Bim[4];
        {
            float r0 = swz16(xr[0]), r1 = swz16(xr[1]), r2 = swz16(xr[2]), r3 = swz16(xr[3]);
            float r4 = swz16(xr[4]), r5 = swz16(xr[5]), r6 = swz16(xr[6]), r7 = swz16(xr[7]);
            v2f t;
            t[0] = lo ? xr[0] : r2;  t[1] = lo ? xr[1] : r3;  Bre[0] = t;  // K=0..3
            t[0] = lo ? xr[4] : r6;  t[1] = lo ? xr[5] : r7;  Bre[1] = t;  // K=4..7
            t[0] = lo ? r0 : xr[2];  t[1] = lo ? r1 : xr[3];  Bre[2] = t;  // K=8..11
            t[0] = lo ? r4 : xr[6];  t[1] = lo ? r5 : xr[7];  Bre[3] = t;  // K=12..15
        }
        {
            float i0 = swz16(xi[0]), i1 = swz16(xi[1]), i2 = swz16(xi[2]), i3 = swz16(xi[3]);
            float i4 = swz16(xi[4]), i5 = swz16(xi[5]), i6 = swz16(xi[6]), i7 = swz16(xi[7]);
            v2f t;
            t[0] = lo ? xi[0] : i2;  t[1] = lo ? xi[1] : i3;  Bim[0] = t;
            t[0] = lo ? xi[4] : i6;  t[1] = lo ? xi[5] : i7;  Bim[1] = t;
            t[0] = lo ? i0 : xi[2];  t[1] = lo ? i1 : xi[3];  Bim[2] = t;
            t[0] = lo ? i4 : xi[6];  t[1] = lo ? i5 : xi[7];  Bim[3] = t;
        }
        // Complex matvec: Out_re = W_re*Zre - W_im*Zim ; Out_im = W_re*Zim + W_im*Zre
#pragma unroll
        for (int kc = 0; kc < 4; ++kc) {
            accR = __builtin_amdgcn_wmma_f32_16x16x4_f32(false, Are[kc],  false, Bre[kc], (short)0, accR, false, false);
            accR = __builtin_amdgcn_wmma_f32_16x16x4_f32(false, Anim[kc], false, Bim[kc], (short)0, accR, false, false);
            accI = __builtin_amdgcn_wmma_f32_16x16x4_f32(false, Are[kc],  false, Bim[kc], (short)0, accI, false, false);
            accI = __builtin_amdgcn_wmma_f32_16x16x4_f32(false, Aim[kc],  false, Bre[kc], (short)0, accI, false, false);
        }
#pragma unroll
        for (int s = 0; s < 8; ++s) { kr[s] = accR[s]; ki[s] = accI[s]; }
    };

    // ---- RK4 3/8-rule, 19 steps
    for (int step = 0; step < 19; ++step) {
        float k1r[8], k1i[8], k2r[8], k2i[8], k3r[8], k3i[8], k4r[8], k4i[8];
        float tr[8], ti[8];

        dyn(zr, zi, k1r, k1i);
#pragma unroll
        for (int s = 0; s < 8; ++s) {
            tr[s] = fmaf(hstep * (1.0f / 3.0f), k1r[s], zr[s]);
            ti[s] = fmaf(hstep * (1.0f / 3.0f), k1i[s], zi[s]);
        }
        dyn(tr, ti, k2r, k2i);
#pragma unroll
        for (int s = 0; s < 8; ++s) {
            tr[s] = fmaf(hstep, k2r[s] - k1r[s] * (1.0f / 3.0f), zr[s]);
            ti[s] = fmaf(hstep, k2i[s] - k1i[s] * (1.0f / 3.0f), zi[s]);
        }
        dyn(tr, ti, k3r, k3i);
#pragma unroll
        for (int s = 0; s < 8; ++s) {
            tr[s] = fmaf(hstep, k1r[s] - k2r[s] + k3r[s], zr[s]);
            ti[s] = fmaf(hstep, k1i[s] - k2i[s] + k3i[s], zi[s]);
        }
        dyn(tr, ti, k4r, k4i);
#pragma unroll
        for (int s = 0; s < 8; ++s) {
            zr[s] = fmaf(hstep * 0.125f, k1r[s] + 3.0f * k2r[s] + 3.0f * k3r[s] + k4r[s], zr[s]);
            zi[s] = fmaf(hstep * 0.125f, k1i[s] + 3.0f * k2i[s] + 3.0f * k3i[s] + k4i[s], zi[s]);
        }
    }

    // ---- Head: out[b] = [re|im] @ head_w.T + head_b   (partial per lane-half, combine via swizzle)
    float acc[4];
#pragma unroll
    for (int o = 0; o < 4; ++o) {
        const float* hw = head_w + o * 32 + 8 * hi;
        float a = 0.0f;
#pragma unroll
        for (int s = 0; s < 8; ++s) {
            a = fmaf(hw[s],      zr[s], a);
            a = fmaf(hw[16 + s], zi[s], a);
        }
        a += swz16(a);          // combine osc 0-7 and 8-15 halves (full EXEC here)
        acc[o] = a + head_b[o];
    }
    if (lo) {
        float4 r = { acc[0], acc[1], acc[2], acc[3] };
        *(float4*)(out + (size_t)(b0 + col) * 4) = r;
    }
}

extern "C" void kernel_launch(void* const* d_in, const int* in_sizes, int n_in,
                              void* d_out, int out_size, void* d_ws, size_t ws_size,
                              hipStream_t stream) {
    const int*   board  = (const int*)d_in[0];
    const float* W_re   = (const float*)d_in[1];
    const float* W_im   = (const float*)d_in[2];
    const float* head_w = (const float*)d_in[3];
    const float* head_b = (const float*)d_in[4];
    float* out = (float*)d_out;

    const int nboards = in_sizes[0] / 16;          // (B, 16) int32
    const int waves   = (nboards + 15) / 16;       // 16 boards per wave32
    const long long threads = (long long)waves * 32;
    const int block = 256;
    const int grid  = (int)((threads + block - 1) / block);

    hopf_ode_head_kernel<<<grid, block, 0, stream>>>(board, W_re, W_im, head_w,
                                                     head_b, out, nboards);
}